// smooth_l1_disparity_and_edge_warp_28432683499919
// MI455X (gfx1250) — compile-verified
//
#include <hip/hip_runtime.h>
#include <math.h>

typedef float v2f __attribute__((ext_vector_type(2)));
typedef float v8f __attribute__((ext_vector_type(8)));

#define WIDTH 960
#define N_TOTAL (16 * 544 * 960)
#define N4 (N_TOTAL / 4)

__device__ __forceinline__ float smooth_l1(float d) {
    float ad = fabsf(d);
    return ad < 1.0f ? 0.5f * d * d : ad - 0.5f;
}

// Exact f32 sum of all 32 lanes' p via V_WMMA_F32_16X16X4_F32.
// A: VGPR0 = p (K0 for lanes 0-15 / K2 for lanes 16-31), VGPR1 = 0.
// B: all ones (4x16).  D[m][n] = r_m = p_m + p_{m+16}.
// D layout: VGPR j holds r_j (lanes 0-15) / r_{j+8} (lanes 16-31), so a local
// 8-register add gives S_low (lanes 0-15) / S_high (lanes 16-31); one
// shfl_xor(16) completes the wave sum. Requires EXEC all-ones (call from
// non-divergent code with full waves).
__device__ __forceinline__ float wave_sum32(float p) {
    v2f a;
    a[0] = p;
    a[1] = 0.0f;
    v2f b;
    b[0] = 1.0f;
    b[1] = 1.0f;
    v8f c = {};
    v8f d = __builtin_amdgcn_wmma_f32_16x16x4_f32(
        /*neg_a=*/false, a, /*neg_b=*/false, b,
        /*c_mod=*/(short)0, c, /*reuse_a=*/false, /*reuse_b=*/false);
    float s = d[0] + d[1] + d[2] + d[3] + d[4] + d[5] + d[6] + d[7];
    s += __shfl_xor(s, 16, 32);
    return s;
}

__global__ void ws_init_kernel(float* ws) {
    if (threadIdx.x < 3) ws[threadIdx.x] = 0.0f;
}

__global__ void __launch_bounds__(256)
loss_main_kernel(const float* __restrict__ dl, const float* __restrict__ seg_r,
                 const float* __restrict__ dlgt, const float* __restrict__ lgt,
                 float* __restrict__ ws) {
    float loss_s = 0.0f;   // smooth_l1(dl - dlgt) * mask
    float recon_s = 0.0f;  // smooth_l1(sigmoid(warp(seg_r, dl)) - lgt) * mask
    float cnt_s = 0.0f;    // mask count

    const int stride = gridDim.x * blockDim.x;
    for (int i = blockIdx.x * blockDim.x + threadIdx.x; i < N4; i += stride) {
        const int e0 = i * 4;             // first flat element of this float4
        const int w0 = e0 % WIDTH;        // WIDTH % 4 == 0 -> same row for all 4
        const int rowStart = e0 - w0;     // row base into seg_r

        const float4 dlv = reinterpret_cast<const float4*>(dl)[i];
        const float4 gtv = reinterpret_cast<const float4*>(dlgt)[i];
        const float4 lv  = reinterpret_cast<const float4*>(lgt)[i];

        const float dla[4] = {dlv.x, dlv.y, dlv.z, dlv.w};
        const float gta[4] = {gtv.x, gtv.y, gtv.z, gtv.w};
        const float lga[4] = {lv.x, lv.y, lv.z, lv.w};

#pragma unroll
        for (int j = 0; j < 4; ++j) {
            const float dval = dla[j];
            const float lval = lga[j];
            const float mf = (lval > 0.0f) ? 1.0f : 0.0f;

            // disparity smooth-L1 term
            loss_s += mf * smooth_l1(dval - gta[j]);

            // horizontal bilinear warp of seg_r at (w - disp), zeros padding
            const float sx = (float)(w0 + j) - dval;
            const float x0 = floorf(sx);
            const float wx = sx - x0;
            const int x0i = (int)x0;
            const int x1i = x0i + 1;
            const float v0 = (x0i >= 0 && x0i < WIDTH) ? 1.0f : 0.0f;
            const float v1 = (x1i >= 0 && x1i < WIDTH) ? 1.0f : 0.0f;
            const int x0c = min(max(x0i, 0), WIDTH - 1);
            const int x1c = min(max(x1i, 0), WIDTH - 1);
            const float g0 = seg_r[rowStart + x0c];
            const float g1 = seg_r[rowStart + x1c];
            const float warped = (1.0f - wx) * g0 * v0 + wx * g1 * v1;
            const float recon = 1.0f / (1.0f + __expf(-warped));

            recon_s += mf * smooth_l1(recon - lval);
            cnt_s += mf;
        }
    }

    // wave32 reduction via WMMA (EXEC all-ones here: no divergence, full waves)
    const float lw = wave_sum32(loss_s);
    const float rw = wave_sum32(recon_s);
    const float cw = wave_sum32(cnt_s);

    __shared__ float sm[3][8];
    const int wave = threadIdx.x >> 5;
    const int lane = threadIdx.x & 31;
    if (lane == 0) {
        sm[0][wave] = lw;
        sm[1][wave] = rw;
        sm[2][wave] = cw;
    }
    __syncthreads();
    if (threadIdx.x == 0) {
        float a = 0.0f, b = 0.0f, c = 0.0f;
#pragma unroll
        for (int k = 0; k < 8; ++k) {
            a += sm[0][k];
            b += sm[1][k];
            c += sm[2][k];
        }
        atomicAdd(&ws[0], a);
        atomicAdd(&ws[1], b);
        atomicAdd(&ws[2], c);
    }
}

__global__ void finalize_kernel(const float* __restrict__ ws,
                                float* __restrict__ out) {
    const float cnt = ws[2];
    float loss = ws[0] / cnt;
    if (loss != loss) loss = 0.0f;  // NaN guard on disparity term only (as ref)
    out[0] = loss + 0.5f * (ws[1] / cnt);
}

extern "C" void kernel_launch(void* const* d_in, const int* in_sizes, int n_in,
                              void* d_out, int out_size, void* d_ws, size_t ws_size,
                              hipStream_t stream) {
    const float* dl    = (const float*)d_in[0];
    const float* seg_r = (const float*)d_in[1];
    const float* dlgt  = (const float*)d_in[2];
    const float* lgt   = (const float*)d_in[3];
    float* ws = (float*)d_ws;
    float* out = (float*)d_out;

    ws_init_kernel<<<1, 32, 0, stream>>>(ws);
    loss_main_kernel<<<1024, 256, 0, stream>>>(dl, seg_r, dlgt, lgt, ws);
    finalize_kernel<<<1, 1, 0, stream>>>(ws, out);
}